// PointsToVolumes_73443940762066
// MI455X (gfx1250) — compile-verified
//
#include <hip/hip_runtime.h>

#define G 256
#define G3 ((size_t)G * (size_t)G * (size_t)G)
#define NCLS 2
#define NB 2

// ---------------------------------------------------------------------------
// Kernel 1: zero-fill the 268MB output volume via the CDNA5 async
// LDS->global store path (GLOBAL_STORE_ASYNC_FROM_LDS_B128, ASYNCcnt).
// A 16B zero block in LDS (only LDS object -> offset 0; init through a
// volatile pointer so the ds_store can't be DCE'd) is streamed out by every
// lane, 4 x b128 per loop trip using immediate offsets to amortize address
// math. Drained with s_wait_asynccnt 0. HBM-bound: ~268MB / 23.3TB/s ~ 11.5us.
// ---------------------------------------------------------------------------
__global__ void pv_zero_async(float* __restrict__ out,
                              unsigned long long n64,      // # of 64B groups
                              unsigned long long tail16) { // leftover 16B chunks
    __shared__ __align__(16) float zbuf[4];  // only LDS allocation -> LDS offset 0
    volatile float* vz = zbuf;               // volatile: keep the ds_store_b32s
    if (threadIdx.x < 4) vz[threadIdx.x] = 0.0f;
    __syncthreads();
    asm volatile("s_wait_dscnt 0" ::: "memory");  // LDS init visible to async engine

    unsigned lds_off = 0u;  // byte offset of zbuf in LDS
    unsigned long long tid    = (unsigned long long)blockIdx.x * blockDim.x + threadIdx.x;
    unsigned long long stride = (unsigned long long)gridDim.x * blockDim.x;

    for (unsigned long long i = tid; i < n64; i += stride) {
        const float* gaddr = out + i * 16ull;  // 64 bytes per group
        asm volatile(
            "global_store_async_from_lds_b128 %0, %1, off\n\t"
            "global_store_async_from_lds_b128 %0, %1, off offset:16\n\t"
            "global_store_async_from_lds_b128 %0, %1, off offset:32\n\t"
            "global_store_async_from_lds_b128 %0, %1, off offset:48"
            :
            : "v"(gaddr), "v"(lds_off)
            : "memory");
    }
    // Tail: remaining 16B chunks (not taken for this problem size).
    if (tid < tail16) {
        const float* gaddr = out + (n64 * 16ull + tid * 4ull);
        asm volatile("global_store_async_from_lds_b128 %0, %1, off"
                     :
                     : "v"(gaddr), "v"(lds_off)
                     : "memory");
    }
    asm volatile("s_wait_asynccnt 0" ::: "memory");
}

// ---------------------------------------------------------------------------
// Kernel 2: one thread per (batch, point). Trilinear weights for the 8
// corners; 2 classes -> 16 native f32 global atomic adds per point
// (global_atomic_add_f32, no-return, STOREcnt). Corners at coordinate G
// (possible since p in [0,G)) are skipped, matching the reference's
// zeroed-weight handling of invalid corners.
// ---------------------------------------------------------------------------
__global__ void pv_scatter(const float* __restrict__ pos,
                           const float* __restrict__ amp,
                           float* __restrict__ out, int N) {
    int tid   = blockIdx.x * blockDim.x + threadIdx.x;
    int total = NB * N;
    if (tid >= total) return;
    int b = tid / N;
    int n = tid - b * N;

    const float* p = pos + (size_t)tid * 3;
    float px = (p[0] + 0.5f) * (float)G;
    float py = (p[1] + 0.5f) * (float)G;
    float pz = (p[2] + 0.5f) * (float)G;
    float fx = floorf(px), fy = floorf(py), fz = floorf(pz);
    int   ix = (int)fx,    iy = (int)fy,    iz = (int)fz;
    float rx = px - fx,    ry = py - fy,    rz = pz - fz;

    float wx[2] = {1.0f - rx, rx};
    float wy[2] = {1.0f - ry, ry};
    float wz[2] = {1.0f - rz, rz};

    float a0 = amp[((size_t)b * NCLS + 0) * N + n];
    float a1 = amp[((size_t)b * NCLS + 1) * N + n];

    float* out0 = out + (size_t)b * NCLS * G3;  // class-0 plane for batch b
    float* out1 = out0 + G3;                    // class-1 plane

#pragma unroll
    for (int dx = 0; dx < 2; ++dx) {
#pragma unroll
        for (int dy = 0; dy < 2; ++dy) {
#pragma unroll
            for (int dz = 0; dz < 2; ++dz) {
                int x = ix + dx, y = iy + dy, z = iz + dz;
                if ((unsigned)x < (unsigned)G && (unsigned)y < (unsigned)G &&
                    (unsigned)z < (unsigned)G) {
                    size_t idx = (((size_t)z * G + (size_t)y) * G + (size_t)x);
                    float  w   = wx[dx] * wy[dy] * wz[dz];
                    unsafeAtomicAdd(out0 + idx, a0 * w);
                    unsafeAtomicAdd(out1 + idx, a1 * w);
                }
            }
        }
    }
}

extern "C" void kernel_launch(void* const* d_in, const int* in_sizes, int n_in,
                              void* d_out, int out_size, void* d_ws, size_t ws_size,
                              hipStream_t stream) {
    const float* positions  = (const float*)d_in[0];  // [B, N, 3] f32
    const float* amplitudes = (const float*)d_in[1];  // [B, C, N] f32
    float*       out        = (float*)d_out;          // [B, C, G, G, G] f32

    int N = in_sizes[0] / (3 * NB);  // B=2 per reference setup

    // out_size = 2*2*256^3 = 67,108,864 floats = 4,194,304 x 64B groups.
    unsigned long long n16    = (unsigned long long)out_size / 4ull;
    unsigned long long n64    = n16 / 4ull;
    unsigned long long tail16 = n16 % 4ull;
    pv_zero_async<<<4096, 256, 0, stream>>>(out, n64, tail16);

    // Scatter (same stream -> ordered after the fill).
    int total  = NB * N;
    int blocks = (total + 255) / 256;
    pv_scatter<<<blocks, 256, 0, stream>>>(positions, amplitudes, out, N);
}